// Yolov2Loss_22290880266923
// MI455X (gfx1250) — compile-verified
//
#include <hip/hip_runtime.h>
#include <hip/hip_bf16.h>

// ---------------------------------------------------------------------------
// YOLOv2 loss for MI455X (gfx1250, wave32).
//
// Roofline: ~170 MB streamed @ 23.3 TB/s => ~7.3 us floor; ~0.5 GFLOP of
// scalar fp32 VALU. Memory/VALU bound; WMMA cannot express IoU/softmax math,
// so matrix cores are used only where they fit: a deterministic fp32
// reduction via V_WMMA_F32_16X16X4_F32 with an all-ones B matrix
// (D[m][n] = sum_k A[m][k] + C[m][n] -> 64 partials folded per instruction).
// ---------------------------------------------------------------------------

namespace {
constexpr int Nn = 64, Aa = 5, Cc = 20, Fm = 52, Mm = 32;
constexpr int PLANE = Fm * Fm;                 // 2704
constexpr int TOTAL = Nn * Aa * PLANE;         // 865280 cells
constexpr int BLK   = 256;
constexpr int NB    = (TOTAL + BLK - 1) / BLK; // 3380 (exact)
constexpr int NBPAD = ((NB + 63) / 64) * 64;   // 3392, multiple of 64 for WMMA tiles
}

typedef float v2f __attribute__((ext_vector_type(2)));
typedef float v8f __attribute__((ext_vector_type(8)));

__device__ __forceinline__ float smooth_l1(float d) {
    float ad = fabsf(d);
    return ad < 1.0f ? 0.5f * d * d : ad - 0.5f;
}

__device__ __forceinline__ float sigmoidf_fast(float x) {
    return 1.0f / (1.0f + __expf(-x));
}

// Zero the padded partial-sum workspace every call (d_ws is poisoned, not
// re-cleared between replays, and the pad tail is never written by main).
__global__ void yolo_zero_ws(float* __restrict__ p, int n) {
    int i = blockIdx.x * BLK + threadIdx.x;
    if (i < n) p[i] = 0.0f;
}

// One thread per (n, a, y, x) cell; consecutive threads map to consecutive x,
// so every strided channel-plane read is a fully coalesced 128B line/wave.
__global__ __launch_bounds__(BLK) void yolo_loss_main(
    const float* __restrict__ preds,      // [N, A*25, FM, FM]
    const float* __restrict__ loc_t,      // [N, A, 4, FM, FM]
    const float* __restrict__ cls_t,      // [N, A, C, FM, FM]
    const float* __restrict__ box_t,      // [N, M, 4]
    const float* __restrict__ anchors,    // [A, 2]
    float* __restrict__ partials)         // SoA: [4][NBPAD]
{
    const int tid = threadIdx.x;
    const int t   = blockIdx.x * BLK + tid;

    float loc_loss = 0.f, iou_loss = 0.f, cls_loss = 0.f, npos = 0.f;

    if (t < TOTAL) {
        const int x  = t % Fm;
        const int y  = (t / Fm) % Fm;
        const int a  = (t / PLANE) % Aa;
        const int n  = t / (PLANE * Aa);
        const int yx = y * Fm + x;

        const float* pp = preds + ((size_t)(n * Aa + a) * (5 + Cc)) * PLANE + yx;
        const float* lt = loc_t + ((size_t)(n * Aa + a) * 4) * PLANE + yx;
        const float* ct = cls_t + ((size_t)(n * Aa + a) * Cc) * PLANE + yx;

        // CDNA5 global_prefetch_b8 on the streaming planes we touch next.
        __builtin_prefetch(pp + 5 * PLANE, 0, 1);
        __builtin_prefetch(ct, 0, 1);

        // ---- raw predictions (each value read exactly once) ----
        const float tx = pp[0 * PLANE], ty = pp[1 * PLANE];
        const float tw = pp[2 * PLANE], th = pp[3 * PLANE];
        const float to = pp[4 * PLANE];
        float logits[Cc];
#pragma unroll
        for (int c = 0; c < Cc; ++c) logits[c] = pp[(5 + c) * PLANE];

        float ctv[Cc];
        float cmax = 0.0f;
#pragma unroll
        for (int c = 0; c < Cc; ++c) { ctv[c] = ct[c * PLANE]; cmax = fmaxf(cmax, ctv[c]); }
        const bool pos = cmax > 0.0f;

        const float sx = sigmoidf_fast(tx);
        const float sy = sigmoidf_fast(ty);
        const float ew = __expf(tw);
        const float eh = __expf(th);

        // ---- localization loss (pos-masked) ----
        if (pos) {
            loc_loss += smooth_l1(sx - lt[0 * PLANE]);
            loc_loss += smooth_l1(sy - lt[1 * PLANE]);
            loc_loss += smooth_l1(ew - lt[2 * PLANE]);
            loc_loss += smooth_l1(eh - lt[3 * PLANE]);
            npos = 1.0f;
        }

        // ---- decode box, max IoU over the 32 GT boxes of this image ----
        const float bx = sx + (float)x;
        const float by = sy + (float)y;
        const float bw = anchors[a * 2 + 0] * ew;
        const float bh = anchors[a * 2 + 1] * eh;
        const float px1 = bx - 0.5f * bw, py1 = by - 0.5f * bh;
        const float px2 = bx + 0.5f * bw, py2 = by + 0.5f * bh;
        const float area_p = bw * bh;

        const float4* bt4 = reinterpret_cast<const float4*>(box_t + (size_t)n * Mm * 4);
        float best = 0.0f;
#pragma unroll 8
        for (int m = 0; m < Mm; ++m) {
            const float4 b = bt4[m];                 // L0/L2 resident (32 KB total)
            float iw = fminf(px2, b.z) - fmaxf(px1, b.x);
            float ih = fminf(py2, b.w) - fmaxf(py1, b.y);
            iw = fmaxf(iw, 0.0f);
            ih = fmaxf(ih, 0.0f);
            const float inter  = iw * ih;
            const float area_t = (b.z - b.x) * (b.w - b.y);
            best = fmaxf(best, inter / (area_p + area_t - inter));
        }

        // ---- objectness/IoU loss ----
        const float ipred = sigmoidf_fast(to);
        const float imask = pos ? 1.0f : 0.1f;
        iou_loss = smooth_l1(ipred * imask - best * imask);

        // ---- softmax + masked class loss ----
        float lmax = logits[0];
#pragma unroll
        for (int c = 1; c < Cc; ++c) lmax = fmaxf(lmax, logits[c]);
        float esum = 0.0f;
        float e[Cc];
#pragma unroll
        for (int c = 0; c < Cc; ++c) { e[c] = __expf(logits[c] - lmax); esum += e[c]; }
        const float rinv = 1.0f / esum;
#pragma unroll
        for (int c = 0; c < Cc; ++c)
            if (ctv[c] > 0.0f) cls_loss += smooth_l1(e[c] * rinv - ctv[c]);
    }

    // ---- deterministic in-block tree reduction (LDS) ----
    __shared__ float sm[4][BLK];
    sm[0][tid] = loc_loss;
    sm[1][tid] = iou_loss;
    sm[2][tid] = cls_loss;
    sm[3][tid] = npos;
    __syncthreads();
#pragma unroll
    for (int off = BLK / 2; off > 0; off >>= 1) {
        if (tid < off) {
            sm[0][tid] += sm[0][tid + off];
            sm[1][tid] += sm[1][tid + off];
            sm[2][tid] += sm[2][tid + off];
            sm[3][tid] += sm[3][tid + off];
        }
        __syncthreads();
    }
    if (tid == 0) {
        partials[0 * NBPAD + blockIdx.x] = sm[0][0];
        partials[1 * NBPAD + blockIdx.x] = sm[1][0];
        partials[2 * NBPAD + blockIdx.x] = sm[2][0];
        partials[3 * NBPAD + blockIdx.x] = sm[3][0];
    }
}

// Final reduction: 4 waves, one per quantity. Each wave folds 64 fp32
// partials per V_WMMA_F32_16X16X4_F32 (B = all-ones => D[m][n] = sum_k A[m][k]
// + C[m][n]; every A element contributes once to every column, so the lane->
// (m,k) mapping is irrelevant for the grand total). Column 0 of the 16x16
// accumulator lives in lanes 0 and 16 (rows 0-7 / 8-15 across the 8 C VGPRs).
__global__ __launch_bounds__(128) void yolo_loss_final(
    const float* __restrict__ partials, float* __restrict__ out)
{
    const int wave = threadIdx.x >> 5;   // quantity: 0=loc 1=iou 2=cls 3=npos
    const int lane = threadIdx.x & 31;
    const float* src = partials + (size_t)wave * NBPAD;

    float total;
#if __has_builtin(__builtin_amdgcn_wmma_f32_16x16x4_f32)
    v8f acc = {};
    v2f ones; ones.x = 1.0f; ones.y = 1.0f;      // B(4x16) = all ones
    for (int i = 0; i < NBPAD; i += 64) {
        v2f aa;
        aa.x = src[i + 2 * lane];
        aa.y = src[i + 2 * lane + 1];
        // 8 args: (neg_a, A, neg_b, B, c_mod, C, reuse_a, reuse_b)
        acc = __builtin_amdgcn_wmma_f32_16x16x4_f32(
            false, aa, false, ones, (short)0, acc, false, false);
    }
    float s = acc[0] + acc[1] + acc[2] + acc[3] + acc[4] + acc[5] + acc[6] + acc[7];
    // column-n total = s(lane n) + s(lane n+16); all columns identical.
    total = __shfl(s, 0, 32) + __shfl(s, 16, 32);
#else
    float s = 0.0f;
    for (int i = lane; i < NBPAD; i += 32) s += src[i];
    for (int off = 16; off > 0; off >>= 1) s += __shfl_down(s, off, 32);
    total = __shfl(s, 0, 32);
#endif

    __shared__ float totals[4];
    if (lane == 0) totals[wave] = total;
    __syncthreads();
    if (threadIdx.x == 0)
        out[0] = (totals[0] + totals[1] + totals[2]) / totals[3];
}

extern "C" void kernel_launch(void* const* d_in, const int* in_sizes, int n_in,
                              void* d_out, int out_size, void* d_ws, size_t ws_size,
                              hipStream_t stream) {
    (void)in_sizes; (void)n_in; (void)out_size; (void)ws_size;
    const float* preds   = (const float*)d_in[0];
    const float* loc_t   = (const float*)d_in[1];
    const float* cls_t   = (const float*)d_in[2];
    const float* box_t   = (const float*)d_in[3];
    const float* anchors = (const float*)d_in[4];
    float* out = (float*)d_out;
    float* ws  = (float*)d_ws;       // needs 4*NBPAD*4 = ~54 KB

    yolo_zero_ws<<<(4 * NBPAD + BLK - 1) / BLK, BLK, 0, stream>>>(ws, 4 * NBPAD);
    yolo_loss_main<<<NB, BLK, 0, stream>>>(preds, loc_t, cls_t, box_t, anchors, ws);
    yolo_loss_final<<<1, 128, 0, stream>>>(ws, out);
}